// GCN_11141145166375
// MI455X (gfx1250) — compile-verified
//
#include <hip/hip_runtime.h>
#include <hip/hip_bf16.h>

// ---------------------------------------------------------------------------
// GCN layer: agg = scatter_add(x[edge_src], edge_dst);  h = agg @ W + b
//   x:   [N, 64] f32      edge_src/edge_dst: [E] int     W: [64,64] f32  b: [64]
// Phase 1: zero agg scratch (d_ws)
// Phase 2: edge scatter — 1 edge per wave32, float2 per lane, f32 L2 atomics
// Phase 3: GEMM via V_WMMA_F32_16X16X4_F32 — 16x16 tile per wave, K=64
// ---------------------------------------------------------------------------

typedef __attribute__((ext_vector_type(2))) float v2f;
typedef __attribute__((ext_vector_type(8))) float v8f;

#define FDIM 64

// ---------------- Phase 1: zero the aggregation buffer ----------------------
__global__ __launch_bounds__(256) void gcn_zero_kernel(float4* __restrict__ p, int n4) {
    int i = blockIdx.x * blockDim.x + threadIdx.x;
    if (i < n4) {
        p[i] = make_float4(0.f, 0.f, 0.f, 0.f);
    }
}

// ---------------- Phase 2: gather rows of x, scatter-add into agg -----------
// One edge per wave32: wave-uniform index loads (scalarizable), each lane
// moves a float2 (full 256B row per wave), two f32 atomics per lane.
// agg (25.6MB) and x (25.6MB) both fit in the 192MB L2 -> atomics stay on-chip.
__global__ __launch_bounds__(256) void gcn_scatter_kernel(
    const float* __restrict__ x,
    const int*   __restrict__ edge_src,
    const int*   __restrict__ edge_dst,
    float*       __restrict__ agg,
    int n_edges)
{
    const int wid  = threadIdx.x >> 5;          // wave id in block (0..7), wave-uniform
    const int lane = threadIdx.x & 31;
    const int edge = blockIdx.x * 8 + wid;      // wave-uniform -> s_load of indices
    if (edge >= n_edges) return;

    const int s = edge_src[edge];
    const int d = edge_dst[edge];

    const float2 v = *(const float2*)(x + (size_t)s * FDIM + lane * 2);
    float* p = agg + (size_t)d * FDIM + lane * 2;
    atomicAdd(p,     v.x);   // global_atomic_add_f32 (no return)
    atomicAdd(p + 1, v.y);
}

// ---------------- Phase 3: h = agg @ W + b via f32 WMMA ---------------------
// One wave computes one 16x16 output tile. K = 64 -> 16 x V_WMMA_F32_16X16X4_F32.
//
// VGPR layouts (ISA 7.12.2, wave32):
//   A 16x4 f32 : lane L, vgpr v  -> A[M = L%16, K = 2*(L/16) + v]   (float2 load)
//   B 4x16 f32 : lane L, vgpr v  -> B[K = v + 2*(L/16), N = L%16]
//   C/D 16x16  : lane L, vgpr v  -> C[M = v + 8*(L/16), N = L%16]
__global__ __launch_bounds__(256) void gcn_gemm_wmma_kernel(
    const float* __restrict__ agg,
    const float* __restrict__ W,
    const float* __restrict__ bias,
    float*       __restrict__ out,
    int n_nodes)
{
    const int tiles_n = FDIM / 16;              // 4 tiles across the 64 columns
    const int wave    = blockIdx.x * 8 + (threadIdx.x >> 5);
    const int tile_m  = wave / tiles_n;
    const int tile_n  = wave % tiles_n;
    const int row0    = tile_m * 16;
    if (row0 >= n_nodes) return;                // whole wave exits together

    const int lane = threadIdx.x & 31;
    const int half = lane >> 4;                 // 0: lanes 0-15, 1: lanes 16-31
    const int lr   = lane & 15;
    const int col0 = tile_n * 16;

    // Accumulator initialized with bias (every C element in column N gets b[N]).
    const float bv = bias[col0 + lr];
    v8f acc;
#pragma unroll
    for (int v = 0; v < 8; ++v) acc[v] = bv;

    // Clamp A-row for (theoretical) ragged tail; 100000 % 16 == 0 so normally exact.
    const int arow = (row0 + lr < n_nodes) ? (row0 + lr) : (n_nodes - 1);
    const float* arow_p = agg + (size_t)arow * FDIM;

#pragma unroll
    for (int k0 = 0; k0 < FDIM; k0 += 4) {
        v2f a, bm;
        // A: two consecutive K values per lane -> single float2 load
        const float2 av = *(const float2*)(arow_p + k0 + 2 * half);
        a[0] = av.x;
        a[1] = av.y;
        // B: two rows of W, coalesced across the 16 lanes of each half
        bm[0] = W[(k0 + 0 + 2 * half) * FDIM + col0 + lr];
        bm[1] = W[(k0 + 1 + 2 * half) * FDIM + col0 + lr];

        // v_wmma_f32_16x16x4_f32  (8-arg form: neg_a, A, neg_b, B, c_mod, C, reuse_a, reuse_b)
        acc = __builtin_amdgcn_wmma_f32_16x16x4_f32(
            false, a, false, bm, (short)0, acc, false, false);
    }

    // Store D tile (divergent guard only AFTER all WMMAs -> EXEC all-ones during WMMA)
#pragma unroll
    for (int v = 0; v < 8; ++v) {
        const int rm = row0 + v + 8 * half;
        if (rm < n_nodes) {
            out[(size_t)rm * FDIM + col0 + lr] = acc[v];
        }
    }
}

// ---------------------------------------------------------------------------
extern "C" void kernel_launch(void* const* d_in, const int* in_sizes, int n_in,
                              void* d_out, int out_size, void* d_ws, size_t ws_size,
                              hipStream_t stream) {
    const float* x        = (const float*)d_in[0];
    const int*   edge_src = (const int*)d_in[1];
    const int*   edge_dst = (const int*)d_in[2];
    const float* W        = (const float*)d_in[3];
    const float* bias     = (const float*)d_in[4];
    float*       out      = (float*)d_out;

    const int n_nodes = in_sizes[0] / FDIM;     // 100000
    const int n_edges = in_sizes[1];            // 1250000

    float* agg = (float*)d_ws;                  // N * 64 f32 = 25.6 MB scratch

    // Phase 1: zero agg
    {
        const int n4     = n_nodes * (FDIM / 4);
        const int blocks = (n4 + 255) / 256;
        gcn_zero_kernel<<<blocks, 256, 0, stream>>>((float4*)agg, n4);
    }

    // Phase 2: edge scatter (1 edge per wave, 8 waves per block)
    {
        const int blocks = (n_edges + 7) / 8;
        gcn_scatter_kernel<<<blocks, 256, 0, stream>>>(x, edge_src, edge_dst, agg, n_edges);
    }

    // Phase 3: WMMA GEMM (16x16 tile per wave, 8 waves per block)
    {
        const int tiles_m = (n_nodes + 15) / 16;
        const int waves   = tiles_m * (FDIM / 16);
        const int blocks  = (waves + 7) / 8;
        gcn_gemm_wmma_kernel<<<blocks, 256, 0, stream>>>(agg, W, bias, out, n_nodes);
    }
}